// PT_MAP_79551384256626
// MI455X (gfx1250) — compile-verified
//
#include <hip/hip_runtime.h>
#include <hip/hip_bf16.h>

#define DEVINL __device__ __forceinline__

typedef float v2f __attribute__((ext_vector_type(2)));
typedef float v8f __attribute__((ext_vector_type(8)));

namespace {
constexpr int   WAY     = 64;
constexpr int   SHOT    = 5;
constexpr int   QUERY   = 195;
constexpr int   DIM     = 1536;
constexpr int   N_ALL   = WAY * (SHOT + QUERY);   // 12800
constexpr int   NLAB    = WAY * SHOT;             // 320
constexpr int   NUNL    = N_ALL - NLAB;           // 12480
constexpr float LAM     = 10.0f;
constexpr float ALPHA   = 0.2f;
constexpr int   EPOCHS  = 20;
constexpr int   MAXITERS = 1000;                  // reference Sinkhorn cap
constexpr float EPS     = 1e-6f;                  // reference convergence tol
constexpr float LOGC    = 5.2729995815f;          // log(195) = column marginal
constexpr float E_M1    = 1.71828182845904523536f;// e - 1
constexpr float MASKSUM_LAB = 315.0f + 5.0f * 2.71828182845904523536f; // 315 + 5e
constexpr int   MTILES  = NUNL / 16;              // 780 WMMA M-tiles (dist GEMM)
constexpr int   KSPLIT  = 16;                     // split-K for the M-step GEMM
constexpr int   KCHUNK  = NUNL / KSPLIT;          // 780 (multiple of 4)
constexpr int   SBLOCKS = 64;                     // persistent Sinkhorn grid
constexpr int   SWAVES  = SBLOCKS * 8;            // 512 waves
}

// ---------------- reductions (deterministic, fixed order) ----------------

DEVINL float wred_sum(float v) {
#pragma unroll
  for (int o = 16; o; o >>= 1) v += __shfl_xor(v, o, 32);
  return v;
}
DEVINL float block_sum_256(float v) {
  __shared__ float sm[8];
  __shared__ float res;
  const int lane = threadIdx.x & 31, wid = threadIdx.x >> 5;
  v = wred_sum(v);
  if (lane == 0) sm[wid] = v;
  __syncthreads();
  if (wid == 0) {
    float r = (lane < 8) ? sm[lane] : 0.0f;
    r = wred_sum(r);
    if (lane == 0) res = r;
  }
  __syncthreads();
  return res;
}

// ---------------- CDNA5 WMMA: D = A(16x4,f32) * B(4x16,f32) + C(16x16,f32) ----
// A lane layout: m = lane&15, VGPR0/1 = K = (lane>>4)*2 + {0,1}
// B lane layout: n = lane&15, VGPR0/1 = K = (lane>>4)*2 + {0,1}
// C lane layout: n = lane&15, VGPR r -> M = r + (lane>>4)*8

DEVINL v8f wmma4(v2f a, v2f b, v8f c) {
  return __builtin_amdgcn_wmma_f32_16x16x4_f32(false, a, false, b, (short)0, c,
                                               false, false);
}

// ---------------- preprocessing ----------------

__global__ void k_rownorm(const float* __restrict__ X, float* __restrict__ Z) {
  const int row = blockIdx.x;
  const float* x = X + (size_t)row * DIM;
  float* z = Z + (size_t)row * DIM;
  float loc[6];
  float s = 0.0f;
#pragma unroll
  for (int j = 0; j < 6; j++) {
    float v = sqrtf(x[threadIdx.x + j * 256] + 1e-6f);
    loc[j] = v;
    s += v * v;
  }
  s = block_sum_256(s);
  const float inv = 1.0f / fmaxf(sqrtf(s), 1e-12f);
#pragma unroll
  for (int j = 0; j < 6; j++) z[threadIdx.x + j * 256] = loc[j] * inv;
}

__global__ void k_colmeans(const float* __restrict__ Z, float* __restrict__ meanL,
                           float* __restrict__ meanU) {
  const int col = blockIdx.x * 256 + threadIdx.x;
  float sl = 0.0f, su = 0.0f;
  for (int i = 0; i < NLAB; i++) sl += Z[(size_t)i * DIM + col];
  for (int i = NLAB; i < N_ALL; i++) su += Z[(size_t)i * DIM + col];
  meanL[col] = sl / (float)NLAB;
  meanU[col] = su / (float)NUNL;
}

__global__ void k_center(float* __restrict__ Z, const float* __restrict__ meanL,
                         const float* __restrict__ meanU) {
  const int row = blockIdx.x;
  const float* mean = (row < NLAB) ? meanL : meanU;
  float* z = Z + (size_t)row * DIM;
  float loc[6];
  float s = 0.0f;
#pragma unroll
  for (int j = 0; j < 6; j++) {
    const int c = threadIdx.x + j * 256;
    float v = z[c] - mean[c];
    loc[j] = v;
    s += v * v;
  }
  s = block_sum_256(s);
  const float inv = 1.0f / fmaxf(sqrtf(s), 1e-12f);
#pragma unroll
  for (int j = 0; j < 6; j++) z[threadIdx.x + j * 256] = loc[j] * inv;
}

__global__ void k_colsumL(const float* __restrict__ Z, float* __restrict__ csl) {
  const int col = blockIdx.x * 256 + threadIdx.x;
  float s = 0.0f;
  for (int i = 0; i < NLAB; i++) s += Z[(size_t)i * DIM + col];
  csl[col] = s;
}

// mus init from shots; constant labeled M-step part: emus_lab = csl + (e-1)*5*mus0
__global__ void k_initmus(const float* __restrict__ Z, const float* __restrict__ csl,
                          float* __restrict__ mus, float* __restrict__ emus_lab) {
  const int w = blockIdx.x;
  const int d = blockIdx.y * 256 + threadIdx.x;
  float s = 0.0f;
#pragma unroll
  for (int sh = 0; sh < SHOT; sh++) s += Z[(size_t)(sh * WAY + w) * DIM + d];
  mus[(size_t)w * DIM + d] = s / (float)SHOT;
  emus_lab[(size_t)w * DIM + d] = csl[d] + E_M1 * s;
}

__global__ void k_musT(const float* __restrict__ mus, float* __restrict__ musT,
                       float* __restrict__ musn2) {
  const int w = blockIdx.x;
  float acc = 0.0f;
#pragma unroll
  for (int j = 0; j < 6; j++) {
    const int d = threadIdx.x + j * 256;
    const float v = mus[(size_t)w * DIM + d];
    musT[(size_t)d * WAY + w] = v;
    acc += v * v;
  }
  acc = block_sum_256(acc);
  if (threadIdx.x == 0) musn2[w] = acc;
}

// ---------------- dist GEMM (WMMA): one wave = one 16x16 tile; 3120 waves ------
__global__ void k_dist(const float* __restrict__ Z, const float* __restrict__ musT,
                       const float* __restrict__ musn2, float* __restrict__ lp0,
                       float* __restrict__ lp0T) {
  const int gw = (int)((blockIdx.x * blockDim.x + threadIdx.x) >> 5);
  if (gw >= MTILES * 4) return;  // whole-wave exit keeps EXEC all-ones for WMMA
  const int lane = threadIdx.x & 31;
  const int mtile = gw >> 2;
  const int n0 = (gw & 3) * 16;
  const int i0 = mtile * 16;
  const int m = lane & 15;
  const int koff = (lane >> 4) * 2;
  const float* arow = Z + (size_t)(NLAB + i0 + m) * DIM + koff;
  const float* bcol = musT + (size_t)koff * WAY + n0 + m;
  v8f acc = {};
  for (int k = 0; k < DIM; k += 4) {
    const v2f a = *(const v2f*)(arow + k);
    const float* bp = bcol + (size_t)k * WAY;
    const v2f b = {bp[0], bp[WAY]};
    acc = wmma4(a, b, acc);
  }
  const int rofs = (lane >> 4) * 8;
  const int w = n0 + m;
  const float mn2 = 1.0f + musn2[w];  // ||Z_u|| == 1 by construction
#pragma unroll
  for (int r = 0; r < 8; ++r) {
    const int i = i0 + r + rofs;
    const float lp = -LAM * sqrtf(fmaxf(mn2 - 2.0f * acc[r], 0.0f));
    lp0[(size_t)i * WAY + w] = lp;
    lp0T[(size_t)w * NUNL + i] = lp;
  }
}

// ---------------- persistent Sinkhorn (single kernel, grid barrier) ------------

DEVINL void grid_bar(int* bar, int& phase) {
  __syncthreads();
  ++phase;
  if (threadIdx.x == 0) {
    __threadfence();
    __hip_atomic_fetch_add(bar, 1, __ATOMIC_RELEASE, __HIP_MEMORY_SCOPE_AGENT);
    const int target = phase * SBLOCKS;
    while (__hip_atomic_load(bar, __ATOMIC_ACQUIRE, __HIP_MEMORY_SCOPE_AGENT) < target)
      __builtin_amdgcn_s_sleep(2);
  }
  __syncthreads();
}

// Exact log-domain Sinkhorn while-loop (row marginal 1, col marginal 195):
//   f_new = -lse_row(lp0+g);  rows = exp(f_old + lse_row(lp0+g))  [= reference lr]
//   stop when max|rows - u_prev| <= EPS (checked BEFORE committing f), cap 1000
//   g = log(c) - lse_col(lp0 + f)
__global__ void k_sinkhorn(const float* __restrict__ lp0, const float* __restrict__ lp0T,
                           float* __restrict__ f, float* __restrict__ g,
                           float* __restrict__ fnew, float* __restrict__ uprev,
                           unsigned* __restrict__ diff, int* __restrict__ bar) {
  const int lane = threadIdx.x & 31;
  const int wid = threadIdx.x >> 5;
  const int gwave = blockIdx.x * 8 + wid;
  __shared__ float sm[256], ss[256];
  __shared__ unsigned sdiff[8];
  __shared__ float sdmax;
  int phase = 0;

  for (int it = 0; it < MAXITERS; ++it) {
    // ---- row phase (one wave per row, 2 cols per lane)
    const float g0 = g[lane];
    const float g1 = g[lane + 32];
    float wdiff = 0.0f;
    for (int row = gwave; row < NUNL; row += SWAVES) {
      const float x0 = lp0[(size_t)row * WAY + lane] + g0;
      const float x1 = lp0[(size_t)row * WAY + lane + 32] + g1;
      float mx = fmaxf(x0, x1);
#pragma unroll
      for (int o = 16; o; o >>= 1) mx = fmaxf(mx, __shfl_xor(mx, o, 32));
      float s = expf(x0 - mx) + expf(x1 - mx);
#pragma unroll
      for (int o = 16; o; o >>= 1) s += __shfl_xor(s, o, 32);
      const float t = mx + logf(s);
      if (lane == 0) {
        fnew[row] = -t;
        const float rowsum = expf(f[row] + t);
        wdiff = fmaxf(wdiff, fabsf(rowsum - uprev[row]));
        uprev[row] = rowsum;
      }
    }
    if (lane == 0) sdiff[wid] = __float_as_uint(wdiff);  // wdiff >= 0
    __syncthreads();
    if (threadIdx.x == 0) {
      unsigned bmax = 0;
      for (int i = 0; i < 8; ++i) bmax = (sdiff[i] > bmax) ? sdiff[i] : bmax;
      atomicMax(&diff[it & 1], bmax);
    }
    grid_bar(bar, phase);
    if (threadIdx.x == 0)
      sdmax = __uint_as_float(
          __hip_atomic_load(&diff[it & 1], __ATOMIC_ACQUIRE, __HIP_MEMORY_SCOPE_AGENT));
    __syncthreads();
    if (it > 0 && sdmax <= EPS) break;  // grid-uniform exit; f,g untouched
    if (blockIdx.x == 0 && threadIdx.x == 0) diff[(it + 1) & 1] = 0u;
    // ---- commit f
    for (int row = gwave; row < NUNL; row += SWAVES)
      if (lane == 0) f[row] = fnew[row];
    grid_bar(bar, phase);
    // ---- col phase (one column per block, online logsumexp)
    {
      const int w = blockIdx.x;
      const float* colp = lp0T + (size_t)w * NUNL;
      float m = -1e30f, s = 0.0f;
      for (int i = threadIdx.x; i < NUNL; i += 256) {
        const float x = colp[i] + f[i];
        if (x > m) { s = s * expf(m - x) + 1.0f; m = x; }
        else       { s += expf(x - m); }
      }
      sm[threadIdx.x] = m;
      ss[threadIdx.x] = s;
      __syncthreads();
      for (int o = 128; o; o >>= 1) {
        if (threadIdx.x < o) {
          const float m1 = sm[threadIdx.x], s1 = ss[threadIdx.x];
          const float m2 = sm[threadIdx.x + o], s2 = ss[threadIdx.x + o];
          const float M = fmaxf(m1, m2);
          sm[threadIdx.x] = M;
          ss[threadIdx.x] = s1 * expf(m1 - M) + s2 * expf(m2 - M);
        }
        __syncthreads();
      }
      if (threadIdx.x == 0) g[w] = LOGC - (sm[0] + logf(ss[0]));
    }
    grid_bar(bar, phase);
  }
}

// ---------------- M-step ----------------

// maskT[w][i] = exp(lp0 + f + g) and its column sum, fused; one block per way
__global__ void k_mask_colsum(const float* __restrict__ lp0T, const float* __restrict__ f,
                              const float* __restrict__ g, float* __restrict__ maskT,
                              float* __restrict__ colsum) {
  const int w = blockIdx.x;
  const float gw = g[w];
  const float* lp = lp0T + (size_t)w * NUNL;
  float* mk = maskT + (size_t)w * NUNL;
  float s = 0.0f;
  for (int i = threadIdx.x; i < NUNL; i += 256) {
    const float v = expf(lp[i] + f[i] + gw);
    mk[i] = v;
    s += v;
  }
  s = block_sum_256(s);
  if (threadIdx.x == 0) colsum[w] = MASKSUM_LAB + s;
}

// split-K partial GEMM: part[s] = maskT[:, ks:ks+KCHUNK] @ Z_u[ks:ks+KCHUNK, :]
// one wave = one 16-way M-tile x 4 dim N-tiles x one K chunk; 1536 waves
__global__ void k_emus_part(const float* __restrict__ Z, const float* __restrict__ maskT,
                            float* __restrict__ part) {
  const int gw = (int)((blockIdx.x * blockDim.x + threadIdx.x) >> 5);
  if (gw >= 96 * KSPLIT) return;
  const int lane = threadIdx.x & 31;
  const int split = gw / 96;
  const int wv = gw % 96;
  const int mtile = wv / 24;      // 0..3 (which 16 ways)
  const int n0 = (wv % 24) * 64;  // 4 N-tiles of 16 dims
  const int m = lane & 15;
  const int koff = (lane >> 4) * 2;
  const float* ap = maskT + (size_t)(mtile * 16 + m) * NUNL + koff;
  const int k0 = split * KCHUNK, k1 = k0 + KCHUNK;
  v8f acc[4] = {};
  for (int k = k0; k < k1; k += 4) {
    __builtin_prefetch(ap + k + 256);
    const v2f a = *(const v2f*)(ap + k);
    const float* bp = Z + (size_t)(NLAB + k + koff) * DIM + n0 + m;
    const v2f b0 = {bp[0], bp[DIM]};
    const v2f b1 = {bp[16], bp[DIM + 16]};
    const v2f b2 = {bp[32], bp[DIM + 32]};
    const v2f b3 = {bp[48], bp[DIM + 48]};
    acc[0] = wmma4(a, b0, acc[0]);
    acc[1] = wmma4(a, b1, acc[1]);
    acc[2] = wmma4(a, b2, acc[2]);
    acc[3] = wmma4(a, b3, acc[3]);
  }
  float* pb = part + (size_t)split * WAY * DIM;
  const int rofs = (lane >> 4) * 8;
#pragma unroll
  for (int t = 0; t < 4; t++) {
#pragma unroll
    for (int r = 0; r < 8; r++) {
      const int w = mtile * 16 + r + rofs;
      const int d = n0 + t * 16 + m;
      pb[(size_t)w * DIM + d] = acc[t][r];
    }
  }
}

// deterministic fixed-order reduction over K splits + prototype update
__global__ void k_emus_red(const float* __restrict__ part, const float* __restrict__ emus_lab,
                           const float* __restrict__ colsum, float* __restrict__ mus) {
  const int e = blockIdx.x * 256 + threadIdx.x;  // < WAY*DIM
  float s = emus_lab[e];
#pragma unroll
  for (int p = 0; p < KSPLIT; ++p) s += part[(size_t)p * WAY * DIM + e];
  const float em = s / colsum[e / DIM];
  mus[e] = mus[e] + ALPHA * (em - mus[e]);
}

// ---------------- outputs ----------------

__global__ void k_zero(int* __restrict__ p, int n) {
  const int e = blockIdx.x * 256 + threadIdx.x;
  if (e < n) p[e] = 0;
}

__global__ void k_outlab(const int* __restrict__ labels, float* __restrict__ out) {
  const int e = blockIdx.x * 256 + threadIdx.x;  // < NLAB*WAY
  const int i = e >> 6, w = e & 63;
  out[e] = (labels[i] == w) ? 1.0f : 0.0f;
}

__global__ void k_outunl(const float* __restrict__ lp0, const float* __restrict__ f,
                         const float* __restrict__ g, const int* __restrict__ labels,
                         float* __restrict__ out, int* __restrict__ cnt) {
  const int u = (int)((blockIdx.x * blockDim.x + threadIdx.x) >> 5);
  if (u >= NUNL) return;
  const int lane = threadIdx.x & 31;
  const float fv = f[u];
  const float x0 = lp0[(size_t)u * WAY + lane] + fv + g[lane];
  const float x1 = lp0[(size_t)u * WAY + lane + 32] + fv + g[lane + 32];
  const size_t ob = (size_t)(NLAB + u) * WAY;
  out[ob + lane] = x0;
  out[ob + lane + 32] = x1;
  float bv;
  int bi;
  if (x1 > x0) { bv = x1; bi = lane + 32; } else { bv = x0; bi = lane; }
#pragma unroll
  for (int o = 16; o; o >>= 1) {
    const float ov = __shfl_xor(bv, o, 32);
    const int oi = __shfl_xor(bi, o, 32);
    if (ov > bv || (ov == bv && oi < bi)) { bv = ov; bi = oi; }
  }
  if (lane == 0 && bi == labels[NLAB + u]) atomicAdd(cnt, 1);  // int atomic: deterministic
}

__global__ void k_finalize(const int* __restrict__ cnt, float* __restrict__ out) {
  out[(size_t)N_ALL * WAY] = (float)(*cnt) / (float)NUNL;  // acc
  out[(size_t)N_ALL * WAY + 1] = 0.0f;                     // pm = std(scalar)*1.96 = 0
}

// ---------------- host orchestration ----------------

extern "C" void kernel_launch(void* const* d_in, const int* in_sizes, int n_in,
                              void* d_out, int out_size, void* d_ws, size_t ws_size,
                              hipStream_t stream) {
  (void)in_sizes; (void)n_in; (void)out_size; (void)ws_size;
  const float* X = (const float*)d_in[0];
  const int* labels = (const int*)d_in[1];
  float* out = (float*)d_out;
  float* ws = (float*)d_ws;

  size_t o = 0;
  float* Z        = ws + o; o += (size_t)N_ALL * DIM;
  float* mus      = ws + o; o += (size_t)WAY * DIM;
  float* musT     = ws + o; o += (size_t)DIM * WAY;
  float* musn2    = ws + o; o += WAY;
  float* emus_lab = ws + o; o += (size_t)WAY * DIM;
  float* csl      = ws + o; o += DIM;
  float* meanL    = ws + o; o += DIM;
  float* meanU    = ws + o; o += DIM;
  float* lp0      = ws + o; o += (size_t)NUNL * WAY;
  float* lp0T     = ws + o; o += (size_t)WAY * NUNL;
  float* maskT    = ws + o; o += (size_t)WAY * NUNL;
  float* colsum   = ws + o; o += WAY;
  float* part     = ws + o; o += (size_t)KSPLIT * WAY * DIM;
  float* fnew     = ws + o; o += NUNL;
  // contiguous zero-init region (f, g, u_prev, diff slots, barrier, match counter)
  float* zbase    = ws + o;
  float* fvec     = ws + o; o += NUNL;
  float* gvec     = ws + o; o += WAY;
  float* uprev    = ws + o; o += NUNL;
  unsigned* diff  = (unsigned*)(ws + o); o += 2;
  int* bar        = (int*)(ws + o); o += 1;
  int* cnt        = (int*)(ws + o); o += 1;
  const int ZN    = NUNL + WAY + NUNL + 2 + 1 + 1;  // 25028 words

  // preprocessing
  k_rownorm<<<N_ALL, 256, 0, stream>>>(X, Z);
  k_colmeans<<<DIM / 256, 256, 0, stream>>>(Z, meanL, meanU);
  k_center<<<N_ALL, 256, 0, stream>>>(Z, meanL, meanU);
  k_colsumL<<<DIM / 256, 256, 0, stream>>>(Z, csl);
  k_initmus<<<dim3(WAY, DIM / 256), 256, 0, stream>>>(Z, csl, mus, emus_lab);

  const int distBlocks = (MTILES * 4) / 8;       // 390 blocks (3120 waves)
  const int zeroBlocks = (ZN + 255) / 256;       // 98
  const int partBlocks = (96 * KSPLIT) / 8;      // 192 blocks (1536 waves)

  for (int ep = 0; ep < EPOCHS; ++ep) {
    k_musT<<<WAY, 256, 0, stream>>>(mus, musT, musn2);
    k_dist<<<distBlocks, 256, 0, stream>>>(Z, musT, musn2, lp0, lp0T);
    k_zero<<<zeroBlocks, 256, 0, stream>>>((int*)zbase, ZN);
    k_sinkhorn<<<SBLOCKS, 256, 0, stream>>>(lp0, lp0T, fvec, gvec, fnew, uprev, diff, bar);
    k_mask_colsum<<<WAY, 256, 0, stream>>>(lp0T, fvec, gvec, maskT, colsum);
    k_emus_part<<<partBlocks, 256, 0, stream>>>(Z, maskT, part);
    k_emus_red<<<(WAY * DIM) / 256, 256, 0, stream>>>(part, emus_lab, colsum, mus);
  }

  // final probas + accuracy
  k_musT<<<WAY, 256, 0, stream>>>(mus, musT, musn2);
  k_dist<<<distBlocks, 256, 0, stream>>>(Z, musT, musn2, lp0, lp0T);
  k_zero<<<zeroBlocks, 256, 0, stream>>>((int*)zbase, ZN);
  k_sinkhorn<<<SBLOCKS, 256, 0, stream>>>(lp0, lp0T, fvec, gvec, fnew, uprev, diff, bar);
  k_outlab<<<(NLAB * WAY) / 256, 256, 0, stream>>>(labels, out);
  k_outunl<<<NUNL / 8, 256, 0, stream>>>(lp0, fvec, gvec, labels, out, cnt);
  k_finalize<<<1, 1, 0, stream>>>(cnt, out);
}